// CustomModule_2121713845124
// MI455X (gfx1250) — compile-verified
//
#include <hip/hip_runtime.h>
#include <hip/hip_bf16.h>
#include <math.h>

// ---------------------------------------------------------------------------
// ANI-1x on MI455X (gfx1250, wave32).
//  - aev_kernel : symmetry functions (radial + angular), fp32 VALU + LDS atomics
//  - mlp_kernel : per-species MLP via V_WMMA_F32_16X16X4_F32 (fp32 matrix pipe)
//  - reduce_kernel : species dispatch + SAE + per-molecule sum
// ---------------------------------------------------------------------------

#define B_MOL 16
#define N_ATOM 48
#define N_SPEC 4
#define AEV_DIM 384
#define PI_F 3.14159265358979f

typedef __attribute__((ext_vector_type(2))) float v2f;
typedef __attribute__((ext_vector_type(8))) float v8f;

// ========================== AEV kernel =====================================
// grid: (48 centers, 16 molecules), block: 64 threads (2 waves)
__global__ void aev_kernel(const int* __restrict__ species,
                           const float* __restrict__ coords,
                           float* __restrict__ aev_out) {
  const int i = blockIdx.x;   // center atom
  const int b = blockIdx.y;   // molecule
  const int tid = threadIdx.x;

  __shared__ float cx[N_ATOM], cy[N_ATOM], cz[N_ATOM];
  __shared__ int sp[N_ATOM];
  __shared__ float dd[N_ATOM], ux[N_ATOM], uy[N_ATOM], uz[N_ATOM];
  __shared__ float acc[AEV_DIM];

  if (tid < N_ATOM) {
    const float* c = coords + (size_t)(b * N_ATOM + tid) * 3;
    cx[tid] = c[0]; cy[tid] = c[1]; cz[tid] = c[2];
    sp[tid] = species[b * N_ATOM + tid];
  }
  for (int f = tid; f < AEV_DIM; f += 64) acc[f] = 0.0f;
  __syncthreads();

  if (tid < N_ATOM) {
    float rx = cx[tid] - cx[i];
    float ry = cy[tid] - cy[i];
    float rz = cz[tid] - cz[i];
    float d2 = rx * rx + ry * ry + rz * rz;
    float d = (tid == i) ? 1.0e6f : sqrtf(d2);
    dd[tid] = d;
    float inv = 1.0f / d;
    ux[tid] = rx * inv; uy[tid] = ry * inv; uz[tid] = rz * inv;
  }
  __syncthreads();

  // ---- radial AEV: features [0,64) = spec*16 + shift ----
  for (int j = tid; j < N_ATOM; j += 64) {
    if (j == i) continue;
    float d = dd[j];
    if (d < 5.2f) {
      float fc = 0.5f * __cosf(PI_F * d * (1.0f / 5.2f)) + 0.5f;
      int base = sp[j] * 16;
#pragma unroll
      for (int t = 0; t < 16; t++) {
        float diff = d - (0.9f + 0.26875f * (float)t);
        float g = 0.25f * __expf(-16.0f * diff * diff) * fc;
        atomicAdd(&acc[base + t], g);
      }
    }
  }

  // ---- angular AEV: features [64,384) = 64 + pairIdx*32 + z*4 + a ----
  int cnt = 0;
  for (int j = 0; j < N_ATOM - 1; j++) {
    for (int k = j + 1; k < N_ATOM; k++) {
      int my = ((cnt++) & 63) == tid;
      if (!my) continue;
      if (j == i || k == i) continue;
      float dj = dd[j], dk = dd[k];
      if (dj >= 3.5f || dk >= 3.5f) continue;
      float fcj = 0.5f * __cosf(PI_F * dj * (1.0f / 3.5f)) + 0.5f;
      float fck = 0.5f * __cosf(PI_F * dk * (1.0f / 3.5f)) + 0.5f;
      float fprod = fcj * fck;
      float cv = ux[j] * ux[k] + uy[j] * uy[k] + uz[j] * uz[k];
      cv = fminf(1.0f, fmaxf(-1.0f, cv));
      float theta = acosf(0.95f * cv);
      float davg = 0.5f * (dj + dk);
      float f2v[4];
#pragma unroll
      for (int a = 0; a < 4; a++) {
        float shfa = 0.9f + 0.65f * (float)a;  // 0.9, 1.55, 2.2, 2.85
        float dv = davg - shfa;
        f2v[a] = __expf(-8.0f * dv * dv);
      }
      int s0 = sp[j], s1 = sp[k];
      int lo = min(s0, s1), hi = max(s0, s1);
      int pidx = lo * 4 - (lo * (lo - 1)) / 2 + (hi - lo);
      int fb = 64 + pidx * 32;
#pragma unroll
      for (int z = 0; z < 8; z++) {
        float shz = (2.0f * (float)z + 1.0f) * (PI_F / 16.0f);
        float x = (1.0f + __cosf(theta - shz)) * 0.5f;
        float x2 = x * x, x4 = x2 * x2, x8 = x4 * x4, x16 = x8 * x8;
        float f1 = x16 * x16;  // x^32
        float t1 = 2.0f * f1 * fprod;
#pragma unroll
        for (int a = 0; a < 4; a++)
          atomicAdd(&acc[fb + z * 4 + a], t1 * f2v[a]);
      }
    }
  }
  __syncthreads();

  float* out = aev_out + (size_t)(b * N_ATOM + i) * AEV_DIM;
  for (int f = tid; f < AEV_DIM; f += 64) out[f] = acc[f];
}

// ========================== MLP kernel =====================================
// One wave per 16-atom tile. fp32 WMMA 16x16x4.
__device__ __forceinline__ float celu01(float x) {
  // CELU with alpha = 0.1
  float neg = 0.1f * (__expf(x * 10.0f) - 1.0f);
  return (x > 0.0f) ? x : neg;
}

// X: LDS [16 x K] row-major. W: global [K x Nout] row-major. Y: LDS [16 x Nout].
__device__ __forceinline__ void mlp_layer(const float* X, int K,
                                          const float* __restrict__ W,
                                          const float* __restrict__ bias,
                                          int Nout, float* Y, bool act,
                                          int lane) {
  const int half = lane >> 4;   // 0: rows K=0,1 ; 1: rows K=2,3
  const int l = lane & 15;
  for (int nn = 0; nn < Nout; nn += 16) {
    v8f accv = {};
    for (int kk = 0; kk < K; kk += 4) {
      int kr = kk + half * 2;
      v2f a, bf;
      a.x = X[l * K + kr];
      a.y = X[l * K + kr + 1];
      bf.x = W[kr * Nout + nn + l];
      bf.y = W[(kr + 1) * Nout + nn + l];
      accv = __builtin_amdgcn_wmma_f32_16x16x4_f32(
          false, a, false, bf, (short)0, accv, false, false);
    }
    float bv = bias[nn + l];
#pragma unroll
    for (int v = 0; v < 8; v++) {
      float x = accv[v] + bv;
      if (act) x = celu01(x);
      int m = v + half * 8;
      Y[m * Nout + nn + l] = x;
    }
  }
}

__global__ void mlp_kernel(const float* __restrict__ aev,
                           const float* __restrict__ W0, const float* __restrict__ b0,
                           const float* __restrict__ W1, const float* __restrict__ b1,
                           const float* __restrict__ W2, const float* __restrict__ b2,
                           const float* __restrict__ W3, const float* __restrict__ b3,
                           int N0, int N1, int N2,
                           float* __restrict__ eat_out) {
  const int tile = blockIdx.x;          // 48 tiles of 16 atoms
  const int lane = threadIdx.x;         // 32 lanes, one wave

  __shared__ float Xs[16 * AEV_DIM];    // 24 KB
  __shared__ float Ys[16 * 160];        // layer0 out / layer2 out (reuse)
  __shared__ float Zs[16 * 128];        // layer1 out

  // stage AEV tile into LDS
  const float* src = aev + (size_t)tile * 16 * AEV_DIM;
  for (int idx = lane; idx < 16 * AEV_DIM; idx += 32) Xs[idx] = src[idx];
  __builtin_amdgcn_s_barrier_signal(-1);
  __builtin_amdgcn_s_barrier_wait(-1);

  mlp_layer(Xs, AEV_DIM, W0, b0, N0, Ys, true, lane);   // 384 -> N0
  mlp_layer(Ys, N0,      W1, b1, N1, Zs, true, lane);   // N0  -> N1
  mlp_layer(Zs, N1,      W2, b2, N2, Ys, true, lane);   // N1  -> N2 (=96)

  // head: N2 -> 1, lanes 0..15 each handle one atom row
  if (lane < 16) {
    float sum = b3[0];
    for (int k = 0; k < N2; k++) sum += Ys[lane * N2 + k] * W3[k];
    eat_out[tile * 16 + lane] = sum;
  }
}

// ========================== reduction ======================================
__global__ void reduce_kernel(const int* __restrict__ species,
                              const float* __restrict__ eat,  // [4][768]
                              const float* __restrict__ sae,
                              float* __restrict__ out) {
  const int b = blockIdx.x;     // molecule
  const int t = threadIdx.x;    // 64 threads
  __shared__ float buf[N_ATOM];
  if (t < N_ATOM) {
    int atom = b * N_ATOM + t;
    int sp = species[atom];
    buf[t] = eat[sp * (B_MOL * N_ATOM) + atom] + sae[sp];
  }
  __syncthreads();
  if (t == 0) {
    float e = 0.0f;
    for (int a = 0; a < N_ATOM; a++) e += buf[a];
    out[b] = e;
  }
}

// ========================== host launcher ==================================
extern "C" void kernel_launch(void* const* d_in, const int* in_sizes, int n_in,
                              void* d_out, int out_size, void* d_ws, size_t ws_size,
                              hipStream_t stream) {
  const int* species = (const int*)d_in[0];
  const float* coords = (const float*)d_in[1];
  // params flatten: for s in 0..3, for layer in 0..3: W, b  -> d_in[2 .. 33]
  const float* sae = (const float*)d_in[34];

  float* aev = (float*)d_ws;                        // [768][384]
  float* eat = aev + (size_t)B_MOL * N_ATOM * AEV_DIM;  // [4][768]

  aev_kernel<<<dim3(N_ATOM, B_MOL), 64, 0, stream>>>(species, coords, aev);

  static const int hdims[4][3] = {
      {160, 128, 96}, {144, 112, 96}, {128, 112, 96}, {128, 112, 96}};
  for (int s = 0; s < 4; s++) {
    const float* W0 = (const float*)d_in[2 + s * 8 + 0];
    const float* b0 = (const float*)d_in[2 + s * 8 + 1];
    const float* W1 = (const float*)d_in[2 + s * 8 + 2];
    const float* b1 = (const float*)d_in[2 + s * 8 + 3];
    const float* W2 = (const float*)d_in[2 + s * 8 + 4];
    const float* b2 = (const float*)d_in[2 + s * 8 + 5];
    const float* W3 = (const float*)d_in[2 + s * 8 + 6];
    const float* b3 = (const float*)d_in[2 + s * 8 + 7];
    mlp_kernel<<<48, 32, 0, stream>>>(aev, W0, b0, W1, b1, W2, b2, W3, b3,
                                      hdims[s][0], hdims[s][1], hdims[s][2],
                                      eat + (size_t)s * B_MOL * N_ATOM);
  }

  reduce_kernel<<<B_MOL, 64, 0, stream>>>(species, eat, sae, (float*)d_out);
}